// HeuristicFindTopNPostprocessing_17162689315249
// MI455X (gfx1250) — compile-verified
//
#include <hip/hip_runtime.h>
#include <hip/hip_bf16.h>
#include <stddef.h>

// ---------------------------------------------------------------------------
// HeuristicFindTopN postprocessing for MI455X (gfx1250, wave32).
//
// Stage 1 (bandwidth-bound, 256 MB stream @ 23.3 TB/s ~= 11 us floor):
//   - async global->LDS double-buffered streaming (ASYNCcnt path)
//   - per-timestep max/argmax over C=64 via lane-pair split + shfl_xor(16)
//   - softmax denominator via chained V_WMMA_F32_16X16X4_F32 with B=ones
//     (A layout per ISA: lanes 0-15 K=0,1 ; lanes 16-31 K=2,3)
//   -> conf[b][t] = 1/sum(exp(x - xmax)), pred[b][t] = argmax  (8 MB to ws)
// Stage 2 (tiny, L2-resident): per-sample run-length encode + iterative
//   top-8 extraction with (value desc, index asc) tiebreak, pad with 0.
// ---------------------------------------------------------------------------

#define B_    128
#define T_    8192
#define C_    64
#define ROWS  64            // timesteps per tile
#define PITCH 68            // floats per LDS row (64 + 4 pad -> conflict-free b64 reads)
#define NT    16            // tiles per block
#define CHUNK (ROWS * NT)   // 1024 timesteps per block
#define TPB1  128           // 4 waves; each wave owns 16 rows of the tile

typedef __attribute__((ext_vector_type(2))) float v2f;
typedef __attribute__((ext_vector_type(4))) float v4f;
typedef __attribute__((ext_vector_type(8))) float v8f;
typedef int v4i_vs __attribute__((vector_size(16)));   // matches builtin's V4i

#if defined(__has_builtin)
#  if __has_builtin(__builtin_amdgcn_global_load_async_to_lds_b128)
#    define HAVE_ASYNC_B128 1
#  endif
#  if __has_builtin(__builtin_amdgcn_s_wait_asynccnt)
#    define HAVE_WAIT_ASYNC 1
#  endif
#  if __has_builtin(__builtin_amdgcn_wmma_f32_16x16x4_f32)
#    define HAVE_WMMA_F32_K4 1
#  endif
#  if __has_builtin(__builtin_amdgcn_wave_barrier)
#    define HAVE_WAVE_BARRIER 1
#  endif
#endif

static __device__ __forceinline__ void wait_async0() {
#if defined(HAVE_WAIT_ASYNC)
  __builtin_amdgcn_s_wait_asynccnt(0);
#elif defined(__AMDGCN__)
  asm volatile("s_wait_asynccnt 0" ::: "memory");
#endif
}

static __device__ __forceinline__ void async_copy16(const float* gp, float* lp) {
#if defined(HAVE_ASYNC_B128)
  __builtin_amdgcn_global_load_async_to_lds_b128(
      (__attribute__((address_space(1))) v4i_vs*)gp,   // global src (AS1)
      (__attribute__((address_space(3))) v4i_vs*)lp,   // LDS dst (AS3)
      /*imm offset*/ 0, /*cpol*/ 0);
#else
  *(v4f*)lp = *(const v4f*)gp;
#endif
}

static __device__ __forceinline__ void load_tile(const float* xb, int tile,
                                                 float* lbuf, int tid) {
  const float* gbase = xb + (size_t)tile * ROWS * C_;
#pragma unroll
  for (int i = 0; i < (ROWS * 16) / TPB1; ++i) {   // 1024 b128 chunks / 128 thr = 8
    int idx = i * TPB1 + tid;
    int r = idx >> 4;        // tile row (timestep)
    int c = idx & 15;        // 16-byte chunk within row
    async_copy16(gbase + r * C_ + c * 4, lbuf + r * PITCH + c * 4);
  }
}

__global__ __launch_bounds__(TPB1)
void stage1_softmax_maxarg(const float* __restrict__ x,
                           float* __restrict__ conf,
                           int*   __restrict__ pred) {
  __shared__ float buf[2][ROWS * PITCH];   // 2 * 64 * 68 * 4 = 34816 B
  __shared__ float sums[TPB1 / 32][16];

  const int tid  = threadIdx.x;
  const int wave = tid >> 5;
  const int lane = tid & 31;
  const int m    = lane & 15;    // row-in-wave (timestep), column of D
  const int hi   = lane >> 4;    // 0: classes {4k,4k+1}, 1: classes {4k+2,4k+3}

  const int chunks_per_sample = T_ / CHUNK;              // 8
  const int b  = blockIdx.x / chunks_per_sample;
  const int t0 = (blockIdx.x % chunks_per_sample) * CHUNK;

  const float* xb    = x + ((size_t)b * T_ + t0) * C_;
  float*       confb = conf + (size_t)b * T_ + t0;
  int*         predb = pred + (size_t)b * T_ + t0;

  load_tile(xb, 0, &buf[0][0], tid);

  for (int tile = 0; tile < NT; ++tile) {
    wait_async0();
    __syncthreads();                         // tile's data visible to all waves
    if (tile + 1 < NT)
      load_tile(xb, tile + 1, &buf[(tile + 1) & 1][0], tid);  // overlap DMA

    const float* rowp = &buf[tile & 1][(wave * 16 + m) * PITCH];

    // ---- pass A: fetch this lane's 32 classes (pairs), local max/argmax ----
    v2f v[16];
#pragma unroll
    for (int kb = 0; kb < 16; ++kb)
      v[kb] = *(const v2f*)(rowp + 4 * kb + 2 * hi);   // ds_load_b64, no bank conflicts

    float bm = -__builtin_huge_valf();
    int   bc = 0;
#pragma unroll
    for (int kb = 0; kb < 16; ++kb) {
      int c0 = 4 * kb + 2 * hi;
      if (v[kb].x > bm) { bm = v[kb].x; bc = c0; }
      if (v[kb].y > bm) { bm = v[kb].y; bc = c0 + 1; }
    }
    // combine lane pair (l, l^16); tie -> lower class index (argmax semantics)
    float om = __shfl_xor(bm, 16, 32);
    int   oc = __shfl_xor(bc, 16, 32);
    float gm;
    int   gc;
    if (om > bm || (om == bm && oc < bc)) { gm = om; gc = oc; }
    else                                  { gm = bm; gc = bc; }

    // ---- pass B: softmax denominator S = sum_c exp(x_c - gm) ----
    float S;
#if defined(HAVE_WMMA_F32_K4)
    v8f acc = {0.f, 0.f, 0.f, 0.f, 0.f, 0.f, 0.f, 0.f};
    v2f ones; ones.x = 1.0f; ones.y = 1.0f;
#pragma unroll
    for (int kb = 0; kb < 16; ++kb) {
      v2f a;
      a.x = __expf(v[kb].x - gm);
      a.y = __expf(v[kb].y - gm);
      // D(16x16) += A(16x4) * ones(4x16): every column n holds row-sums
      acc = __builtin_amdgcn_wmma_f32_16x16x4_f32(
          /*neg_a*/ false, a, /*neg_b*/ false, ones,
          /*c_mod*/ (short)0, acc, /*reuse_a*/ false, /*reuse_b*/ false);
    }
    // column 0 lives in lanes 0 (rows 0-7) and 16 (rows 8-15)
    if (m == 0) {
#pragma unroll
      for (int r = 0; r < 8; ++r) sums[wave][hi * 8 + r] = acc[r];
    }
#if defined(HAVE_WAVE_BARRIER)
    __builtin_amdgcn_wave_barrier();
#else
    asm volatile("" ::: "memory");
#endif
    S = sums[wave][m];   // same-wave LDS ops are in-order
#else
    float s = 0.f;
#pragma unroll
    for (int kb = 0; kb < 16; ++kb)
      s += __expf(v[kb].x - gm) + __expf(v[kb].y - gm);
    S = s + __shfl_xor(s, 16, 32);
#endif

    if (hi == 0) {
      int tg = tile * ROWS + wave * 16 + m;
      confb[tg] = 1.0f / S;      // max(softmax) = exp(0)/S
      predb[tg] = gc;
    }
  }
}

// ---------------------------------------------------------------------------
// Stage 2: per-sample run-length voting + iterative top-8 (8 MB, L2-resident)
// ---------------------------------------------------------------------------
__global__ __launch_bounds__(256)
void stage2_rle_topk(const float* __restrict__ conf,
                     const int*   __restrict__ pred,
                     float* __restrict__ out) {
  __shared__ float         voted[T_];   // 32 KB
  __shared__ unsigned char pc[T_];      //  8 KB (classes fit in u8)
  __shared__ float rv[256];
  __shared__ int   ri[256];

  const int b   = blockIdx.x;
  const int tid = threadIdx.x;
  const float* cb = conf + (size_t)b * T_;
  const int*   pb = pred + (size_t)b * T_;
  const float NINF = -__builtin_huge_valf();

  for (int t = tid; t < T_; t += 256) pc[t] = (unsigned char)pb[t];
  __syncthreads();

  // run starts get conf[start] * run_length; everything else -inf
  for (int t = tid; t < T_; t += 256) {
    bool start = (t == 0) || (pc[t] != pc[t - 1]);
    float vv = NINF;
    if (start) {
      int j = t + 1;
      while (j < T_ && pc[j] == pc[t]) ++j;
      vv = cb[t] * (float)(j - t);
    }
    voted[t] = vv;
  }
  __syncthreads();

  // extract top-8 (value desc, index asc on ties — matches lax.top_k)
  for (int k = 0; k < 8; ++k) {
    float bv = NINF;
    int   bi = T_;
    for (int t = tid; t < T_; t += 256) {
      float vv = voted[t];
      if (vv > bv || (vv == bv && t < bi)) { bv = vv; bi = t; }
    }
    rv[tid] = bv; ri[tid] = bi;
    __syncthreads();
    for (int s = 128; s > 0; s >>= 1) {
      if (tid < s) {
        float ov = rv[tid + s];
        int   oi = ri[tid + s];
        if (ov > rv[tid] || (ov == rv[tid] && oi < ri[tid])) {
          rv[tid] = ov; ri[tid] = oi;
        }
      }
      __syncthreads();
    }
    if (tid == 0) {
      float v0 = rv[0];
      int   i0 = ri[0];
      out[(size_t)b * 8 + k] = (v0 != NINF && i0 < T_) ? (float)pc[i0] : 0.0f;
      if (i0 < T_) voted[i0] = NINF;   // remove winner for next round
    }
    __syncthreads();
  }
}

extern "C" void kernel_launch(void* const* d_in, const int* in_sizes, int n_in,
                              void* d_out, int out_size, void* d_ws, size_t ws_size,
                              hipStream_t stream) {
  (void)in_sizes; (void)n_in; (void)out_size; (void)ws_size;
  const float* x = (const float*)d_in[0];
  float* out = (float*)d_out;

  float* conf = (float*)d_ws;                                          // 4 MB
  int*   pred = (int*)((char*)d_ws + (size_t)B_ * T_ * sizeof(float)); // 4 MB

  stage1_softmax_maxarg<<<B_ * (T_ / CHUNK), TPB1, 0, stream>>>(x, conf, pred);
  stage2_rle_topk<<<B_, 256, 0, stream>>>(conf, pred, out);
}